// GCN_28613072126261
// MI455X (gfx1250) — compile-verified
//
#include <hip/hip_runtime.h>

typedef __attribute__((ext_vector_type(2))) float v2f;
typedef __attribute__((ext_vector_type(8))) float v8f;

#define GCN_N_NODES 100000

// ---------------------------------------------------------------------------
// Degree / normalization
// ---------------------------------------------------------------------------
__global__ void __launch_bounds__(256)
deg_init_kernel(float* __restrict__ deg, int n) {
  int i = blockIdx.x * blockDim.x + threadIdx.x;
  if (i < n) deg[i] = 1.0f;  // self-loop contributes 1 to every node's degree
}

__global__ void __launch_bounds__(256)
deg_count_kernel(const int* __restrict__ dst, float* __restrict__ deg, int E) {
  int e = blockIdx.x * blockDim.x + threadIdx.x;
  if (e < E) unsafeAtomicAdd(&deg[dst[e]], 1.0f);  // global_atomic_add_f32
}

__global__ void __launch_bounds__(256)
deg_rsqrt_kernel(float* __restrict__ deg, int n) {
  int i = blockIdx.x * blockDim.x + threadIdx.x;
  if (i < n) deg[i] = rsqrtf(deg[i]);  // deg >= 1 always (self loop)
}

// ---------------------------------------------------------------------------
// WMMA fp32 GEMM:  C[M,N] = op(A)[M,K] @ W[K,N]
//   FUSE_RELU_BIAS: op(A) = relu(A + bias[k])  (bias indexed by K / input feat)
// One wave per 16x16 output tile, K-loop in steps of 4 (V_WMMA_F32_16X16X4_F32).
// A-operand layout: lane<16 -> K pair {0,1}, lane>=16 -> K pair {2,3}, M=lane&15.
// ---------------------------------------------------------------------------
template <bool FUSE_RELU_BIAS>
__global__ void __launch_bounds__(256)
gemm_wmma_kernel(const float* __restrict__ A, const float* __restrict__ W,
                 const float* __restrict__ bias, float* __restrict__ C,
                 int M, int N, int K, int totalTiles) {
  int wave = blockIdx.x * (blockDim.x >> 5) + (threadIdx.x >> 5);
  if (wave >= totalTiles) return;  // wave-uniform: EXEC stays all-ones for WMMA

  int tilesPerRow = N >> 4;
  int tileRow = wave / tilesPerRow;
  int tileCol = wave - tileRow * tilesPerRow;
  int row0 = tileRow << 4;
  int col0 = tileCol << 4;

  int lane = threadIdx.x & 31;
  int half = lane >> 4;  // 0: K offset 0,1 | 1: K offset 2,3
  int l15  = lane & 15;
  int kOff = half * 2;

  const float* arow = A + (size_t)(row0 + l15) * K;
  const float* wcol = W + col0 + l15;

  v8f acc = {};
#pragma unroll 4
  for (int k = 0; k < K; k += 4) {
    float a0 = arow[k + kOff];
    float a1 = arow[k + kOff + 1];
    if (FUSE_RELU_BIAS) {
      a0 = fmaxf(a0 + bias[k + kOff], 0.0f);
      a1 = fmaxf(a1 + bias[k + kOff + 1], 0.0f);
    }
    v2f a; a.x = a0; a.y = a1;
    v2f b;
    b.x = wcol[(size_t)(k + kOff) * N];
    b.y = wcol[(size_t)(k + kOff + 1) * N];
    acc = __builtin_amdgcn_wmma_f32_16x16x4_f32(
        /*neg_a=*/false, a, /*neg_b=*/false, b,
        /*c_mod=*/(short)0, acc, /*reuse_a=*/false, /*reuse_b=*/false);
  }

  // C/D layout: VGPR r -> M = r (lanes 0-15) or M = r+8 (lanes 16-31), N = l15
  float* crow = C + (size_t)(row0 + half * 8) * N + col0 + l15;
#pragma unroll
  for (int r = 0; r < 8; ++r) crow[(size_t)r * N] = acc[r];
}

// ---------------------------------------------------------------------------
// Accumulator init with fused self-loop term (+ optional output bias):
//   out[i, f] = (bias ? bias[f] : 0) + h[i, f] * dinv[i]^2
// ---------------------------------------------------------------------------
template <int F>
__global__ void __launch_bounds__(256)
init_self_kernel(const float* __restrict__ h, const float* __restrict__ dinv,
                 const float* __restrict__ bias, float* __restrict__ out, int total) {
  int idx = blockIdx.x * blockDim.x + threadIdx.x;
  if (idx >= total) return;
  int i = idx / F;          // F is a power-of-two constant -> shift
  int f = idx & (F - 1);
  float di = dinv[i];
  float v = h[idx] * di * di;
  if (bias) v += bias[f];
  out[idx] = v;
}

// ---------------------------------------------------------------------------
// Edge scatter: one wave32 per edge, FPL floats per lane (F = 32*FPL).
//   out[dst[e], :] += h[src[e], :] * dinv[src]*dinv[dst]
// h and out both L2-resident (<= 51 MB each vs 192 MB L2).
// ---------------------------------------------------------------------------
template <int FPL>
__global__ void __launch_bounds__(256)
scatter_kernel(const float* __restrict__ h, const int* __restrict__ src,
               const int* __restrict__ dst, const float* __restrict__ dinv,
               float* __restrict__ out, int E) {
  const int F = FPL * 32;
  int e = (blockIdx.x * blockDim.x + threadIdx.x) >> 5;
  if (e >= E) return;
  int lane = threadIdx.x & 31;
  int s = src[e];
  int d = dst[e];
  float norm = dinv[s] * dinv[d];
  const float* hp = h + (size_t)s * F + lane * FPL;
  float* op = out + (size_t)d * F + lane * FPL;
  if constexpr (FPL == 4) {
    float4 v = *reinterpret_cast<const float4*>(hp);
    unsafeAtomicAdd(op + 0, v.x * norm);
    unsafeAtomicAdd(op + 1, v.y * norm);
    unsafeAtomicAdd(op + 2, v.z * norm);
    unsafeAtomicAdd(op + 3, v.w * norm);
  } else {
    float2 v = *reinterpret_cast<const float2*>(hp);
    unsafeAtomicAdd(op + 0, v.x * norm);
    unsafeAtomicAdd(op + 1, v.y * norm);
  }
}

// ---------------------------------------------------------------------------
// Launcher
// ---------------------------------------------------------------------------
extern "C" void kernel_launch(void* const* d_in, const int* in_sizes, int n_in,
                              void* d_out, int out_size, void* d_ws, size_t ws_size,
                              hipStream_t stream) {
  const float* x  = (const float*)d_in[0];   // [N, 128]
  const int*   ei = (const int*)d_in[1];     // [2, E]
  const float* W1 = (const float*)d_in[2];   // [128, 128]
  const float* b1 = (const float*)d_in[3];   // [128]
  const float* W2 = (const float*)d_in[4];   // [128, 64]
  const float* b2 = (const float*)d_in[5];   // [64]
  float* out = (float*)d_out;                // [N, 64]

  const int Nn = GCN_N_NODES;
  const int E  = in_sizes[1] / 2;
  const int* src = ei;
  const int* dst = ei + E;

  // Workspace layout (floats): dinv[N] | h[N*128] (reused as h2[N*64]) | out1[N*128]
  float* dinv = (float*)d_ws;
  float* h    = dinv + Nn;
  float* out1 = h + (size_t)Nn * 128;
  float* h2   = h;  // h is dead after the layer-1 scatter

  // --- symmetric degree normalization -------------------------------------
  deg_init_kernel<<<(Nn + 255) / 256, 256, 0, stream>>>(dinv, Nn);
  deg_count_kernel<<<(E + 255) / 256, 256, 0, stream>>>(dst, dinv, E);
  deg_rsqrt_kernel<<<(Nn + 255) / 256, 256, 0, stream>>>(dinv, Nn);

  // --- layer 1: h = x @ W1 ------------------------------------------------
  {
    int M = Nn, N = 128, K = 128;
    int tiles = (M >> 4) * (N >> 4);
    gemm_wmma_kernel<false><<<(tiles + 7) / 8, 256, 0, stream>>>(
        x, W1, nullptr, h, M, N, K, tiles);
  }
  // out1 = h * dinv^2 (self-loop term); bias b1 is fused into the GEMM2 load
  init_self_kernel<128><<<((Nn * 128) + 255) / 256, 256, 0, stream>>>(
      h, dinv, nullptr, out1, Nn * 128);
  scatter_kernel<4><<<(E + 7) / 8, 256, 0, stream>>>(h, src, dst, dinv, out1, E);

  // --- layer 2: h2 = relu(out1 + b1) @ W2 ---------------------------------
  {
    int M = Nn, N = 64, K = 128;
    int tiles = (M >> 4) * (N >> 4);
    gemm_wmma_kernel<true><<<(tiles + 7) / 8, 256, 0, stream>>>(
        out1, W2, b1, h2, M, N, K, tiles);
  }
  // out = b2 + h2 * dinv^2, then scatter edges
  init_self_kernel<64><<<((Nn * 64) + 255) / 256, 256, 0, stream>>>(
      h2, dinv, b2, out, Nn * 64);
  scatter_kernel<2><<<(E + 7) / 8, 256, 0, stream>>>(h2, src, dst, dinv, out, E);
}